// GraphConv_18580028523118
// MI455X (gfx1250) — compile-verified
//
#include <hip/hip_runtime.h>

typedef float v8f __attribute__((ext_vector_type(8)));
typedef float v2f __attribute__((ext_vector_type(2)));

#define N_NODES   100000
#define N_EDGES   1600000
#define IN_FEATS  256
#define OUT_FEATS 64
#define N_TILES   (N_NODES / 16)   // 6250, exact

// ---------------------------------------------------------------------------
// k0: zero the output accumulator and the two degree arrays
// ---------------------------------------------------------------------------
__global__ void gcn_init_kernel(float* __restrict__ out, int* __restrict__ degs) {
    int g = blockIdx.x * blockDim.x + threadIdx.x;
    if (g < N_NODES * OUT_FEATS) out[g] = 0.0f;
    if (g < 2 * N_NODES)         degs[g] = 0;
}

// ---------------------------------------------------------------------------
// k1: integer degree histograms (deg_out over src, deg_in over dst)
// ---------------------------------------------------------------------------
__global__ void gcn_degree_kernel(const int* __restrict__ src,
                                  const int* __restrict__ dst,
                                  int* __restrict__ deg_out,
                                  int* __restrict__ deg_in) {
    int e = blockIdx.x * blockDim.x + threadIdx.x;
    if (e >= N_EDGES) return;
    atomicAdd(&deg_out[src[e]], 1);
    atomicAdd(&deg_in[dst[e]], 1);
}

// ---------------------------------------------------------------------------
// k2: h = (feat @ weight) * rsqrt(max(deg_out,1))   via V_WMMA_F32_16X16X4_F32
// One wave handles a 16-row x 64-col tile: 64 K-steps x 4 N-tiles.
// ---------------------------------------------------------------------------
__global__ __launch_bounds__(256) void gcn_gemm_wmma_kernel(
        const float* __restrict__ feat,
        const float* __restrict__ weight,
        const int*   __restrict__ deg_out,
        float*       __restrict__ h) {
    const int lane = threadIdx.x & 31;
    const int wave = threadIdx.x >> 5;
    const int tile = blockIdx.x * 8 + wave;     // wave-uniform guard -> EXEC all-1s
    if (tile >= N_TILES) return;

    const int rbase = tile * 16;
    const int half  = lane >> 4;                // 0: K+0/K+1, 1: K+2/K+3
    const int l16   = lane & 15;

    // A: lane holds feat[rbase+l16][k0 + 2*half + {0,1}] (8B-aligned float2)
    const float* __restrict__ arow = feat + (size_t)(rbase + l16) * IN_FEATS + 2 * half;
    // B: lane holds weight[k0 + 2*half + {0,1}][ntile*16 + l16]
    const float* __restrict__ wcol = weight + (size_t)(2 * half) * OUT_FEATS + l16;

    v8f acc0 = {}, acc1 = {}, acc2 = {}, acc3 = {};

#pragma unroll 4
    for (int kk = 0; kk < IN_FEATS / 4; ++kk) {
        v2f a = *(const v2f*)(arow + kk * 4);
        const float* wk = wcol + (size_t)kk * 4 * OUT_FEATS;
        v2f b0, b1, b2, b3;
        b0.x = wk[0];  b0.y = wk[OUT_FEATS + 0];
        b1.x = wk[16]; b1.y = wk[OUT_FEATS + 16];
        b2.x = wk[32]; b2.y = wk[OUT_FEATS + 32];
        b3.x = wk[48]; b3.y = wk[OUT_FEATS + 48];
        acc0 = __builtin_amdgcn_wmma_f32_16x16x4_f32(false, a, false, b0, (short)0, acc0, false, false);
        acc1 = __builtin_amdgcn_wmma_f32_16x16x4_f32(false, a, false, b1, (short)0, acc1, false, false);
        acc2 = __builtin_amdgcn_wmma_f32_16x16x4_f32(false, a, false, b2, (short)0, acc2, false, false);
        acc3 = __builtin_amdgcn_wmma_f32_16x16x4_f32(false, a, false, b3, (short)0, acc3, false, false);
    }

    // C/D layout: VGPR j -> row rbase + j + 8*half, col = ntile*16 + l16
#pragma unroll
    for (int j = 0; j < 8; ++j) {
        const int row = rbase + j + 8 * half;
        const float nrm = rsqrtf(fmaxf((float)deg_out[row], 1.0f));
        float* __restrict__ hrow = h + (size_t)row * OUT_FEATS + l16;
        hrow[0]  = acc0[j] * nrm;
        hrow[16] = acc1[j] * nrm;
        hrow[32] = acc2[j] * nrm;
        hrow[48] = acc3[j] * nrm;
    }
}

// ---------------------------------------------------------------------------
// k3: edge scatter — one wave per edge, 2 features per lane, f32 atomics
// ---------------------------------------------------------------------------
__global__ __launch_bounds__(256) void gcn_scatter_kernel(
        const float* __restrict__ h,
        const int*   __restrict__ src,
        const int*   __restrict__ dst,
        float*       __restrict__ out) {
    const long long g = (long long)blockIdx.x * blockDim.x + threadIdx.x;
    const int e    = (int)(g >> 5);
    const int lane = (int)(g & 31);
    if (e >= N_EDGES) return;
    const int s = src[e];
    const int d = dst[e];
    const float* __restrict__ hs = h   + (size_t)s * OUT_FEATS;
    float*       __restrict__ od = out + (size_t)d * OUT_FEATS;
    atomicAdd(&od[lane],      hs[lane]);
    atomicAdd(&od[lane + 32], hs[lane + 32]);
}

// ---------------------------------------------------------------------------
// k4: out = relu(out * rsqrt(max(deg_in,1)) + bias)  (in place)
// ---------------------------------------------------------------------------
__global__ void gcn_finalize_kernel(float* __restrict__ out,
                                    const int* __restrict__ deg_in,
                                    const float* __restrict__ bias) {
    int g = blockIdx.x * blockDim.x + threadIdx.x;
    if (g >= N_NODES * OUT_FEATS) return;
    const int row = g >> 6;
    const int col = g & 63;
    const float nrm = rsqrtf(fmaxf((float)deg_in[row], 1.0f));
    const float v = out[g] * nrm + bias[col];
    out[g] = fmaxf(v, 0.0f);
}

// ---------------------------------------------------------------------------
extern "C" void kernel_launch(void* const* d_in, const int* in_sizes, int n_in,
                              void* d_out, int out_size, void* d_ws, size_t ws_size,
                              hipStream_t stream) {
    const float* feat   = (const float*)d_in[0];
    const float* weight = (const float*)d_in[1];
    const float* bias   = (const float*)d_in[2];
    const int*   src    = (const int*)d_in[3];
    const int*   dst    = (const int*)d_in[4];
    float* out = (float*)d_out;

    // workspace: h [N_NODES*64 f32] | deg_out [N_NODES i32] | deg_in [N_NODES i32]
    float* h       = (float*)d_ws;
    int*   degs    = (int*)((char*)d_ws + (size_t)N_NODES * OUT_FEATS * sizeof(float));
    int*   deg_out = degs;
    int*   deg_in  = degs + N_NODES;

    gcn_init_kernel<<<(N_NODES * OUT_FEATS + 255) / 256, 256, 0, stream>>>(out, degs);
    gcn_degree_kernel<<<(N_EDGES + 255) / 256, 256, 0, stream>>>(src, dst, deg_out, deg_in);
    gcn_gemm_wmma_kernel<<<(N_TILES + 7) / 8, 256, 0, stream>>>(feat, weight, deg_out, h);
    gcn_scatter_kernel<<<(int)(((long long)N_EDGES * 32) / 256), 256, 0, stream>>>(h, src, dst, out);
    gcn_finalize_kernel<<<(N_NODES * OUT_FEATS + 255) / 256, 256, 0, stream>>>(out, deg_in, bias);
}